// SqueezeNet_15461882265621
// MI455X (gfx1250) — compile-verified
//
#include <hip/hip_runtime.h>
#include <math.h>

typedef __attribute__((ext_vector_type(2))) float v2f;
typedef __attribute__((ext_vector_type(4))) float v4f;
typedef __attribute__((ext_vector_type(8))) float v8f;

// CDNA5 async global->LDS staging (ASYNCcnt-tracked), with safe fallback.
#if defined(__AMDGCN__) && __has_builtin(__builtin_amdgcn_global_load_async_to_lds_b128) && \
    __has_builtin(__builtin_amdgcn_s_wait_asynccnt)
#define ASYNC_LDS 1
#else
#define ASYNC_LDS 0
#endif

// ---------------------------------------------------------------------------
// Weight quantization: scale = max|w|/127 ; wq = rint(w/scale)*scale
// ---------------------------------------------------------------------------
__global__ __launch_bounds__(256)
void maxabs_kernel(const float* __restrict__ w, int n, float* __restrict__ scale_out) {
    __shared__ float red[256];
    int tid = threadIdx.x;
    float m = 0.f;
    for (int i = tid; i < n; i += 256) m = fmaxf(m, fabsf(w[i]));
    red[tid] = m;
    __syncthreads();
    for (int s = 128; s > 0; s >>= 1) {
        if (tid < s) red[tid] = fmaxf(red[tid], red[tid + s]);
        __syncthreads();
    }
    if (tid == 0) scale_out[0] = red[0] / 127.0f;
}

__global__ __launch_bounds__(256)
void quantize_kernel(const float* __restrict__ w, float* __restrict__ wq, int n,
                     const float* __restrict__ scale) {
    int i = blockIdx.x * 256 + threadIdx.x;
    if (i >= n) return;
    float s = scale[0];
    wq[i] = (s > 0.f) ? rintf(w[i] / s) * s : 0.f;
}

// ---------------------------------------------------------------------------
// LDS-staged implicit-GEMM conv (stride 1; KH = 1 or 3) + bias + ReLU.
// Block = 8 waves: 8 m-tiles x one (NT*16)-channel output strip.
// Weights for the strip are staged into LDS in K-chunks of KC via
// GLOBAL_LOAD_ASYNC_TO_LDS_B128 (ASYNCcnt) and reused by all 8 waves.
// LDS row stride 132 floats: 16B-aligned rows, conflict-free fragment reads.
// No wave-divergent returns: EXEC is all-ones at every WMMA (ISA 7.12 req).
// ---------------------------------------------------------------------------
#define KC 128
#define LSTR (KC + 4)

template <int KH, int NT>
__global__ __launch_bounds__(256)
void conv_lds_wmma_kernel(const float* __restrict__ in, const float* __restrict__ wq,
                          const float* __restrict__ bias, float* __restrict__ out,
                          int Nb, int Cin, int H, int W, int Cout,
                          int pad, int outCtot, int outCoff) {
    const int KHW = KH * KH;
    const int K   = Cin * KHW;       // multiple of 4 for every 1x1 / 3x3 layer here
    const int hw  = H * W;           // stride 1: output spatial == input spatial
    const int M   = Nb * hw;
    const int TN  = (Cout + 15) >> 4;
    const int TG  = (TN + NT - 1) / NT;

    __shared__ __attribute__((aligned(16))) float lds[NT * 16 * LSTR];

    int tg  = blockIdx.x % TG;
    int mb  = blockIdx.x / TG;
    int wid = threadIdx.x >> 5;
    int lane = threadIdx.x & 31;
    int l15 = lane & 15, lh = lane >> 4, kb = lh * 2;
    int m0  = (mb * 8 + wid) << 4;
    int co0 = tg * (NT * 16);

    // --- A-row (activation) decode for this lane ---
    int  mA   = m0 + l15;
    bool mval = (mA < M);
    int  mc   = mval ? mA : 0;
    int  img  = mc / hw;
    int  rem  = mc - img * hw;
    const float* inp   = in + (size_t)img * Cin * hw;
    const float* aBase = inp + rem;            // 1x1 path
    int  oh  = rem / W;
    int  ow  = rem - oh * W;
    int  ih0 = oh - pad;
    int  iw0 = ow - pad;

    v8f acc[NT] = {};

    for (int kc = 0; kc < K; kc += KC) {
        int remk = K - kc;
        if (remk > KC) remk = KC;
        // ---- stage weight strip chunk: NT*16 rows x remk cols ----
        {
            const int QPR = KC / 4;  // quads per row
            for (int idx = threadIdx.x; idx < NT * 16 * QPR; idx += 256) {
                int cl = idx / QPR;
                int kk = (idx - cl * QPR) * 4;
                if (kk < remk) {
                    int co = co0 + cl;
                    float* lp = &lds[cl * LSTR + kk];
                    if (co < Cout) {
                        const float* gp = wq + (size_t)co * K + kc + kk;
#if ASYNC_LDS
                        typedef int v4i_vs __attribute__((vector_size(16)));
                        typedef __attribute__((address_space(1))) v4i_vs* gv4i_p;
                        typedef __attribute__((address_space(3))) v4i_vs* lv4i_p;
                        __builtin_amdgcn_global_load_async_to_lds_b128(
                            (gv4i_p)gp, (lv4i_p)lp, 0, 0);
#else
                        *(v4f*)lp = *(const v4f*)gp;
#endif
                    } else {
                        v4f z = {0.f, 0.f, 0.f, 0.f};
                        *(v4f*)lp = z;
                    }
                }
            }
#if ASYNC_LDS
            __builtin_amdgcn_s_wait_asynccnt(0);
#endif
            __syncthreads();
        }
        // ---- compute over the chunk ----
        for (int k = 0; k < remk; k += 4) {
            int k0 = k + kb;
            int kk = kc + k0;
            // A fragment
            v2f Af;
            if (KH == 1) {
                Af.x = mval ? aBase[(size_t)kk * hw] : 0.f;
                Af.y = mval ? aBase[(size_t)(kk + 1) * hw] : 0.f;
            } else {
                float a01[2] = {0.f, 0.f};
#pragma unroll
                for (int t = 0; t < 2; ++t) {
                    int kt = kk + t;
                    int ci = kt / 9;
                    int r  = kt - ci * 9;
                    int kh = r / 3;
                    int kwv = r - kh * 3;
                    int ih = ih0 + kh, iw = iw0 + kwv;
                    if (mval && ih >= 0 && ih < H && iw >= 0 && iw < W)
                        a01[t] = inp[((size_t)ci * H + ih) * W + iw];
                }
                Af.x = a01[0];
                Af.y = a01[1];
            }
            // B fragments from LDS (bank-conflict-free; pairs 8B aligned)
            v2f Bf[NT];
#pragma unroll
            for (int j = 0; j < NT; ++j) {
                int la = (j * 16 + l15) * LSTR + k0;
                Bf[j].x = lds[la];
                Bf[j].y = lds[la + 1];
            }
            // NT independent WMMAs sharing the A fragment
#pragma unroll
            for (int j = 0; j < NT; ++j)
                acc[j] = __builtin_amdgcn_wmma_f32_16x16x4_f32(
                    false, Af, false, Bf[j], (short)0, acc[j], false, false);
        }
        __syncthreads();
    }

    // --- writeback: predecode the 8 output rows once ---
    int  imgv[8], r2v[8];
    bool okv[8];
#pragma unroll
    for (int v = 0; v < 8; ++v) {
        int m  = m0 + v + lh * 8;
        okv[v] = (m < M);
        int mm = okv[v] ? m : 0;
        imgv[v] = mm / hw;
        r2v[v]  = mm - imgv[v] * hw;
    }
#pragma unroll
    for (int j = 0; j < NT; ++j) {
        int co = co0 + j * 16 + l15;
        if (co >= Cout) continue;
        float bv = bias[co];
#pragma unroll
        for (int v = 0; v < 8; ++v) {
            if (okv[v]) {
                float val = fmaxf(acc[j][v] + bv, 0.f);  // every conv here is ReLU'd
                out[((size_t)imgv[v] * outCtot + outCoff + co) * (size_t)hw + r2v[v]] = val;
            }
        }
    }
}

// ---------------------------------------------------------------------------
// Register-only implicit-GEMM conv for the 7x7/s2 stem (K=147, runs once).
// ---------------------------------------------------------------------------
__global__ __launch_bounds__(256)
void conv7_wmma_kernel(const float* __restrict__ in, const float* __restrict__ wq,
                       const float* __restrict__ bias, float* __restrict__ out,
                       int Nb, int Cin, int H, int W, int Cout,
                       int stride, int OH, int OW, int outCtot) {
    const int NT = 4;
    const int K = Cin * 49;
    const int M = Nb * OH * OW;
    const int TN = (Cout + 15) >> 4;
    const int TG = (TN + NT - 1) / NT;
    const int TM = (M + 15) >> 4;
    int wave = blockIdx.x * (blockDim.x >> 5) + (threadIdx.x >> 5);
    if (wave >= TM * TG) return;  // wave-uniform

    int tg = wave % TG, tm = wave / TG;
    int m0 = tm << 4;
    int lane = threadIdx.x & 31;
    int l15 = lane & 15, lh = lane >> 4, kb = lh * 2;

    int  mA   = m0 + l15;
    bool mval = (mA < M);
    int  mc   = mval ? mA : 0;
    int  ohw  = OH * OW;
    int  img  = mc / ohw;
    int  rem  = mc - img * ohw;
    const float* inp = in + (size_t)img * Cin * H * W;
    int  oh  = rem / OW;
    int  ow  = rem - oh * OW;
    int  ih0 = oh * stride;
    int  iw0 = ow * stride;

    const float* wrow[NT];
    bool nval[NT];
    int  coA[NT];
#pragma unroll
    for (int j = 0; j < NT; ++j) {
        int co  = (tg * NT + j) * 16 + l15;
        coA[j]  = co;
        nval[j] = (co < Cout);
        wrow[j] = wq + (size_t)(nval[j] ? co : 0) * K;
        __builtin_prefetch(wrow[j], 0, 1);
    }

    v8f acc[NT] = {};
    for (int k = 0; k < K; k += 4) {
        int k0 = k + kb;
        float a01[2] = {0.f, 0.f};
#pragma unroll
        for (int t = 0; t < 2; ++t) {
            int kk = k0 + t;
            if (kk < K) {
                int ci = kk / 49;
                int r  = kk - ci * 49;
                int kh = r / 7;
                int kw = r - kh * 7;
                int ih = ih0 + kh, iw = iw0 + kw;
                if (mval && ih < H && iw < W)
                    a01[t] = inp[((size_t)ci * H + ih) * W + iw];
            }
        }
        v2f Af = {a01[0], a01[1]};
        v2f Bf[NT];
#pragma unroll
        for (int j = 0; j < NT; ++j) {
            float b0 = 0.f, b1 = 0.f;
            if (nval[j]) {
                if (k0 < K)     b0 = wrow[j][k0];
                if (k0 + 1 < K) b1 = wrow[j][k0 + 1];
            }
            Bf[j].x = b0;
            Bf[j].y = b1;
        }
#pragma unroll
        for (int j = 0; j < NT; ++j)
            acc[j] = __builtin_amdgcn_wmma_f32_16x16x4_f32(
                false, Af, false, Bf[j], (short)0, acc[j], false, false);
    }

    int  imgv[8], r2v[8];
    bool okv[8];
#pragma unroll
    for (int v = 0; v < 8; ++v) {
        int m  = m0 + v + lh * 8;
        okv[v] = (m < M);
        int mm = okv[v] ? m : 0;
        imgv[v] = mm / ohw;
        r2v[v]  = mm - imgv[v] * ohw;
    }
#pragma unroll
    for (int j = 0; j < NT; ++j) {
        if (!nval[j]) continue;
        float bv = bias[coA[j]];
#pragma unroll
        for (int v = 0; v < 8; ++v) {
            if (okv[v]) {
                float val = fmaxf(acc[j][v] + bv, 0.f);
                out[((size_t)imgv[v] * outCtot + coA[j]) * (size_t)ohw + r2v[v]] = val;
            }
        }
    }
}

// ---------------------------------------------------------------------------
// Ceil-mode 3x3/s2 maxpool (window clamped == -inf padding at bottom/right)
// ---------------------------------------------------------------------------
__global__ __launch_bounds__(256)
void maxpool_kernel(const float* __restrict__ in, float* __restrict__ out,
                    int NC, int H, int W, int OH, int OW) {
    long long idx = (long long)blockIdx.x * 256 + threadIdx.x;
    long long total = (long long)NC * OH * OW;
    if (idx >= total) return;
    int ow = (int)(idx % OW);
    long long t = idx / OW;
    int oh = (int)(t % OH);
    int nc = (int)(t / OH);
    int h0 = oh * 2, w0 = ow * 2;
    int h1 = min(h0 + 3, H), w1 = min(w0 + 3, W);
    const float* p = in + (size_t)nc * H * W;
    float m = -3.402823466e38f;
    for (int ih = h0; ih < h1; ++ih)
        for (int iw = w0; iw < w1; ++iw)
            m = fmaxf(m, p[ih * W + iw]);
    out[idx] = m;
}

// ---------------------------------------------------------------------------
// Global average pool: mean over HW per (n, c)
// ---------------------------------------------------------------------------
__global__ __launch_bounds__(256)
void mean_kernel(const float* __restrict__ in, float* __restrict__ out, int total, int HW) {
    int i = blockIdx.x * 256 + threadIdx.x;
    if (i >= total) return;
    const float* p = in + (size_t)i * HW;
    float s = 0.f;
#pragma unroll 4
    for (int j = 0; j < HW; ++j) s += p[j];
    out[i] = s * (1.0f / (float)HW);
}

// ---------------------------------------------------------------------------
// Host orchestration
// ---------------------------------------------------------------------------
extern "C" void kernel_launch(void* const* d_in, const int* in_sizes, int n_in,
                              void* d_out, int out_size, void* d_ws, size_t ws_size,
                              hipStream_t stream) {
    (void)n_in; (void)out_size; (void)ws_size;
    const int NB = 64;
    static const int FC[8][4] = {{96,16,64,64},{128,16,64,64},{128,32,128,128},{256,32,128,128},
                                 {256,48,192,192},{384,48,192,192},{384,64,256,256},{512,64,256,256}};

    // ---- input index maps: insertion order vs jax-tree (alphabetical) order ----
    int ix, ic1w, ic1b, ifw, ifb;
    int fsw[8], fsb[8], fe1w[8], fe1b[8], fe3w[8], fe3b[8];
    if (in_sizes[0] == 64 * 3 * 224 * 224) {
        ix = 0; ic1w = 1; ic1b = 2; ifw = 3; ifb = 4;
        for (int i = 0; i < 8; ++i) {
            int b = 5 + 6 * i;
            fsw[i] = b; fsb[i] = b + 1; fe1w[i] = b + 2; fe1b[i] = b + 3; fe3w[i] = b + 4; fe3b[i] = b + 5;
        }
    } else {
        ic1b = 0; ic1w = 1; ifb = 2; ifw = 3;
        for (int i = 0; i < 8; ++i) {
            int b = 4 + 6 * i;
            fe1b[i] = b; fe1w[i] = b + 1; fe3b[i] = b + 2; fe3w[i] = b + 3; fsb[i] = b + 4; fsw[i] = b + 5;
        }
        ix = 52;
    }
    auto F = [&](int i) { return (const float*)d_in[i]; };

    // ---- workspace layout ----
    float* ws = (float*)d_ws;
    size_t off = 0;
    auto alloc = [&](size_t n) { float* p = ws + off; off += n; return p; };
    float* scales = alloc(64);
    float* q_c1 = alloc((size_t)96 * 3 * 49);
    float* q_sq[8]; float* q_e1[8]; float* q_e3[8];
    for (int i = 0; i < 8; ++i) {
        q_sq[i] = alloc((size_t)FC[i][1] * FC[i][0]);
        q_e1[i] = alloc((size_t)FC[i][2] * FC[i][1]);
        q_e3[i] = alloc((size_t)FC[i][3] * FC[i][1] * 9);
    }
    float* q_fin = alloc((size_t)1000 * 512);
    float* S = alloc((size_t)NB * 32 * 54 * 54);    // squeeze scratch
    float* A = alloc((size_t)NB * 96 * 109 * 109);  // big ping buffer (292 MB)
    float* B = alloc((size_t)NB * 128 * 54 * 54);   // pong buffer (95.6 MB)

    // ---- quantize all 26 weight tensors ----
    int sidx = 0;
    auto quant = [&](const float* w, size_t n, float* qw) {
        maxabs_kernel<<<1, 256, 0, stream>>>(w, (int)n, scales + sidx);
        quantize_kernel<<<dim3((unsigned)((n + 255) / 256)), 256, 0, stream>>>(w, qw, (int)n, scales + sidx);
        ++sidx;
    };
    quant(F(ic1w), (size_t)96 * 3 * 49, q_c1);
    for (int i = 0; i < 8; ++i) {
        quant(F(fsw[i]), (size_t)FC[i][1] * FC[i][0], q_sq[i]);
        quant(F(fe1w[i]), (size_t)FC[i][2] * FC[i][1], q_e1[i]);
        quant(F(fe3w[i]), (size_t)FC[i][3] * FC[i][1] * 9, q_e3[i]);
    }
    quant(F(ifw), (size_t)1000 * 512, q_fin);

    // ---- conv launcher (stride-1 1x1 / 3x3 via LDS kernel; 7x7 stem separate) ----
    auto conv = [&](const float* in, const float* qw, const float* bias, float* out,
                    int Cin, int Hin, int Cout, int KH, int pad,
                    int outCtot, int outCoff) {
        long long M = (long long)NB * Hin * Hin;
        int TM = (int)((M + 15) / 16);
        int TN = (Cout + 15) / 16;
        int NT = (KH == 1 && TN < 4) ? 1 : 4;
        int TG = (TN + NT - 1) / NT;
        int MB = (TM + 7) / 8;
        unsigned blocks = (unsigned)((long long)MB * TG);
        if (KH == 1) {
            if (NT == 1)
                conv_lds_wmma_kernel<1, 1><<<blocks, 256, 0, stream>>>(in, qw, bias, out, NB, Cin,
                                                                       Hin, Hin, Cout, pad, outCtot, outCoff);
            else
                conv_lds_wmma_kernel<1, 4><<<blocks, 256, 0, stream>>>(in, qw, bias, out, NB, Cin,
                                                                       Hin, Hin, Cout, pad, outCtot, outCoff);
        } else {
            conv_lds_wmma_kernel<3, 4><<<blocks, 256, 0, stream>>>(in, qw, bias, out, NB, Cin,
                                                                   Hin, Hin, Cout, pad, outCtot, outCoff);
        }
    };
    auto pool = [&](const float* in, float* out, int C, int H, int OH) {
        long long total = (long long)NB * C * OH * OH;
        maxpool_kernel<<<(unsigned)((total + 255) / 256), 256, 0, stream>>>(in, out, NB * C, H, H, OH, OH);
    };
    auto fire = [&](int i, const float* in, float* out, int hw) {
        conv(in, q_sq[i], F(fsb[i]), S, FC[i][0], hw, FC[i][1], 1, 0, FC[i][1], 0);
        int ct = FC[i][2] + FC[i][3];
        conv(S, q_e1[i], F(fe1b[i]), out, FC[i][1], hw, FC[i][2], 1, 0, ct, 0);
        conv(S, q_e3[i], F(fe3b[i]), out, FC[i][1], hw, FC[i][3], 3, 1, ct, FC[i][2]);
    };

    // ---- forward pass ----
    // conv1 7x7 s2: 224 -> 109 (register-only WMMA kernel)
    {
        long long M = (long long)NB * 109 * 109;
        int TM = (int)((M + 15) / 16);
        int TG = ((96 + 15) / 16 + 3) / 4;
        long long tiles = (long long)TM * TG;
        unsigned blocks = (unsigned)((tiles + 7) / 8);
        conv7_wmma_kernel<<<blocks, 256, 0, stream>>>(F(ix), q_c1, F(ic1b), A, NB, 3, 224, 224,
                                                      96, 2, 109, 109, 96);
    }
    pool(A, B, 96, 109, 54);
    fire(0, B, A, 54);
    fire(1, A, B, 54);
    fire(2, B, A, 54);
    pool(A, B, 256, 54, 27);
    fire(3, B, A, 27);
    fire(4, A, B, 27);
    fire(5, B, A, 27);
    fire(6, A, B, 27);
    pool(B, A, 512, 27, 13);
    fire(7, A, B, 13);
    conv(B, q_fin, F(ifb), A, 512, 13, 1000, 1, 0, 1000, 0);
    mean_kernel<<<(64 * 1000 + 255) / 256, 256, 0, stream>>>(A, (float*)d_out, 64 * 1000, 13 * 13);
}